// QuasimetricModel_58265526337622
// MI455X (gfx1250) — compile-verified
//
#include <hip/hip_runtime.h>
#include <hip/hip_bf16.h>
#include <stdint.h>

#define AS1 __attribute__((address_space(1)))
#define AS3 __attribute__((address_space(3)))

#ifndef __has_builtin
#define __has_builtin(x) 0
#endif

#define COMPILER_MEMFENCE() asm volatile("" ::: "memory")

// ---------------- CDNA5 async global -> LDS copy (one f32 per lane) --------
__device__ __forceinline__ void async_load_f32(const float* g, float* l) {
#if defined(__HIP_DEVICE_COMPILE__)
#if __has_builtin(__builtin_amdgcn_global_load_async_to_lds_b32)
    // signature (probe-confirmed): (global int* src, local int* dst, imm offset, imm cpol)
    float* gnc = const_cast<float*>(g);
    __builtin_amdgcn_global_load_async_to_lds_b32(
        (AS1 int*)gnc, (AS3 int*)l, 0, 0);
#else
    // VDST = LDS byte offset (low 32 bits of flat LDS address), VADDR = 64-bit global addr
    asm volatile("global_load_async_to_lds_b32 %0, %1, off"
                 :: "v"((uint32_t)(uintptr_t)l), "v"((uint64_t)(uintptr_t)g)
                 : "memory");
#endif
#else
    (void)g; (void)l;
#endif
}

template <int N>
__device__ __forceinline__ void wait_asynccnt() {
#if defined(__HIP_DEVICE_COMPILE__)
#if __has_builtin(__builtin_amdgcn_s_wait_asynccnt)
    __builtin_amdgcn_s_wait_asynccnt(N);
#else
    asm volatile("s_wait_asynccnt %0" :: "i"(N) : "memory");
#endif
#endif
}

__device__ __forceinline__ uint32_t ballot32(bool p) {
#if defined(__HIP_DEVICE_COMPILE__) && __has_builtin(__builtin_amdgcn_ballot_w32)
    return __builtin_amdgcn_ballot_w32(p);
#else
    return (uint32_t)__ballot(p);
#endif
}

// Order-preserving float -> u32 map (total order, -0 < +0; irrelevant here since
// x<y is false for equal values). Branchless.
__device__ __forceinline__ uint32_t ford(float f) {
    uint32_t b = __float_as_uint(f);
    uint32_t m = (uint32_t)(((int32_t)b) >> 31) | 0x80000000u;
    return b ^ m;
}

// ---------------------------------------------------------------------------
// One block (1024 threads = 32 wave32) per batch row. Each wave handles two
// components (64 endpoints each). Exact rank-count sweep replacement for the
// reference's argsort+cumsum.
// ---------------------------------------------------------------------------
__global__ __launch_bounds__(1024, 1)
void iqe_maxmean_kernel(const float* __restrict__ zx,
                        const float* __restrict__ zy,
                        const float* __restrict__ raw_alpha,
                        float* __restrict__ out) {
    constexpr int WAVES  = 32;   // waves per block
    constexpr int HIDDEN = 2048;

    __shared__ float    stage[WAVES][2][64];   // per-wave staging, double-buffered (16 KB)
    __shared__ uint64_t keys[WAVES][64];       // per-wave sort keys (16 KB)
    __shared__ float    redmax[WAVES];
    __shared__ float    redsum[WAVES];

    const int b    = blockIdx.x;
    const int tid  = threadIdx.x;
    const int wave = tid >> 5;
    const int lane = tid & 31;

    const size_t rowbase = (size_t)b * HIDDEN;

    // Issue async global->LDS loads for both components this wave handles.
    // Order: g0.x, g0.y, g1.x, g1.y  (async loads complete in order).
#pragma unroll
    for (int g = 0; g < 2; ++g) {
        const int c = wave * 2 + g;
        const float* px = zx + rowbase + (size_t)c * 32 + lane;
        const float* py = zy + rowbase + (size_t)c * 32 + lane;
        async_load_f32(px, &stage[wave][g][lane]);
        async_load_f32(py, &stage[wave][g][32 + lane]);
    }

    float wmax = -__builtin_inff();
    float wsum = 0.0f;

#pragma unroll
    for (int g = 0; g < 2; ++g) {
        if (g == 0) wait_asynccnt<2>();   // first component's 2 loads done
        else        wait_asynccnt<0>();   // all done
        COMPILER_MEMFENCE();

        const float xv = stage[wave][g][lane];
        const float yv = stage[wave][g][32 + lane];
        const bool valid = xv < yv;
        const uint32_t vm = ballot32(valid);   // uniform SGPR validity mask

        // Total-order keys: (ordered float)<<6 | endpoint index (x: 0..31, y: 32..63)
        const uint64_t kx = ((uint64_t)ford(xv) << 6) | (uint32_t)lane;
        const uint64_t ky = ((uint64_t)ford(yv) << 6) | (uint32_t)(32 + lane);
        keys[wave][lane]      = kx;
        keys[wave][32 + lane] = ky;
        COMPILER_MEMFENCE();   // same-wave DS ops are processed in order

        // C^- for each owned endpoint: sum of signs over strictly-smaller keys.
        // Opening (x_j, valid): s=-1 ; closing (y_j, valid): s=+1 ; else 0.
        int Cx = 0, Cy = 0;
#pragma unroll 4
        for (int j = 0; j < 32; ++j) {            // openings
            if ((vm >> j) & 1u) {                 // uniform scalar branch
                const uint64_t kj = keys[wave][j];        // LDS broadcast read
                Cx -= (int)(kj < kx);
                Cy -= (int)(kj < ky);
            }
        }
#pragma unroll 4
        for (int j = 0; j < 32; ++j) {            // closings
            if ((vm >> j) & 1u) {
                const uint64_t kj = keys[wave][32 + j];
                Cx += (int)(kj < kx);
                Cy += (int)(kj < ky);
            }
        }

        // contribution_i = e_i * ([C^- < 0] - [C^+ < 0]),  C^+ = C^- + s_i
        const int sx = valid ? -1 : 0;
        const int sy = valid ?  1 : 0;
        float contrib = xv * (float)((int)(Cx < 0) - (int)(Cx + sx < 0))
                      + yv * (float)((int)(Cy < 0) - (int)(Cy + sy < 0));

        // Wave reduction -> component union length (identical in all lanes)
#pragma unroll
        for (int off = 16; off > 0; off >>= 1)
            contrib += __shfl_xor(contrib, off, 32);

        wmax = fmaxf(wmax, contrib);
        wsum += contrib;
    }

    if (lane == 0) { redmax[wave] = wmax; redsum[wave] = wsum; }
    __syncthreads();

    if (wave == 0) {
        float m = redmax[lane];
        float s = redsum[lane];
#pragma unroll
        for (int off = 16; off > 0; off >>= 1) {
            m = fmaxf(m, __shfl_xor(m, off, 32));
            s += __shfl_xor(s, off, 32);
        }
        if (lane == 0) {
            const float a = 1.0f / (1.0f + __expf(-raw_alpha[0]));
            out[b] = a * m + (1.0f - a) * (s * (1.0f / 64.0f));
        }
    }
}

extern "C" void kernel_launch(void* const* d_in, const int* in_sizes, int n_in,
                              void* d_out, int out_size, void* d_ws, size_t ws_size,
                              hipStream_t stream) {
    (void)in_sizes; (void)n_in; (void)d_ws; (void)ws_size;
    const float* zx = (const float*)d_in[0];
    const float* zy = (const float*)d_in[1];
    const float* ra = (const float*)d_in[2];
    float* out = (float*)d_out;

    const int B = out_size;            // 32768 rows, one block each
    dim3 grid(B), block(1024);
    hipLaunchKernelGGL(iqe_maxmean_kernel, grid, block, 0, stream, zx, zy, ra, out);
}